// ChamferLoss_652835029199
// MI455X (gfx1250) — compile-verified
//
#include <hip/hip_runtime.h>
#include <hip/hip_bf16.h>
#include <float.h>

typedef __attribute__((ext_vector_type(2))) float v2f;
typedef __attribute__((ext_vector_type(8))) float v8f;

#define FLT_MAX_BITS 0x7F7FFFFF

// ---------------------------------------------------------------------------
// Phase 0: seed the per-point NN-min arrays with +FLT_MAX bits, zero out[0].
// ---------------------------------------------------------------------------
__global__ void chamfer_init(int* __restrict__ minbits, long long n,
                             float* __restrict__ out) {
  long long i = (long long)blockIdx.x * blockDim.x + threadIdx.x;
  if (i == 0) out[0] = 0.0f;
  if (i < n) minbits[i] = FLT_MAX_BITS;
}

// ---------------------------------------------------------------------------
// Phase 1: one workgroup (8 wave32) per (batch, 16-row tile). Each wave
// handles chunks of 4 consecutive 16-column tiles (64 target points).
// One V_WMMA_F32_16X16X4_F32 per 16x16 tile produces squared distances:
//   A row  m = [px, py, pz, 1]
//   B col  n = [-2tx, -2ty, -2tz, t2]
//   C[m][n]  = p2[m]
//   D[m][n]  = p2 + t2 - 2 p.t = |p - t|^2
// Mins are taken on d^2 (sqrt is monotonic; applied in phase 2).
// Column mins: each 16-lane half atomics its partial min (over its 8 M-rows)
// to the same address -> no cross-half shuffle in the hot loop.
// ---------------------------------------------------------------------------
__global__ __launch_bounds__(256) void chamfer_tiles(
    const float* __restrict__ p, const float* __restrict__ t,
    const long long* __restrict__ nums, const long long* __restrict__ dnums,
    int B, int* __restrict__ d1bits, int* __restrict__ d2bits) {
  // Map blockIdx.x -> (batch b, row-tile rt); compute batch offsets.
  int rem = blockIdx.x;
  int b = -1, rt = 0;
  long long poff = 0, toff = 0, Nb = 0, Mb = 0;
  for (int i = 0; i < B; ++i) {
    long long Ni = nums[i];
    int ti = (int)((Ni + 15) >> 4);
    if (rem < ti) { b = i; rt = rem; Nb = Ni; Mb = dnums[i]; break; }
    rem -= ti;
    poff += Ni;
    toff += dnums[i];
  }
  if (b < 0) return;  // padding block

  // Stage the 16 pred points (x,y,z,p2) of this row tile in LDS.
  __shared__ float sp[16][4];
  const int tid = threadIdx.x;
  if (tid < 16) {
    long long m = (long long)rt * 16 + tid;
    float x, y, z;
    if (m < Nb) {
      const float* pp = p + (poff + m) * 3;
      x = pp[0]; y = pp[1]; z = pp[2];
    } else {
      x = y = z = 1.0e18f;  // pad rows far away; d^2 ~ 3e36, never a min
    }
    sp[tid][0] = x; sp[tid][1] = y; sp[tid][2] = z;
    sp[tid][3] = x * x + y * y + z * z;
  }
  __syncthreads();

  const int lane = tid & 31;
  const int wave = tid >> 5;
  const int hi   = lane >> 4;  // 0: K0/K1 (and M 0..7 in C/D), 1: K2/K3 (M 8..15)
  const int lm   = lane & 15;

  // A fragment: row M = lm, K = [px,py,pz,1]. Lanes 0-15 carry K0,K1;
  // lanes 16-31 carry K2,K3 (32-bit 16x4 A layout).
  v2f a;
  a.x = hi ? sp[lm][2] : sp[lm][0];
  a.y = hi ? 1.0f      : sp[lm][1];

  // C = p2[M] broadcast along N. C/D layout: VGPR r -> M = r + 8*hi.
  v8f c;
#pragma unroll
  for (int r = 0; r < 8; ++r) c[r] = sp[r + 8 * hi][3];

  float rowacc[8];
#pragma unroll
  for (int r = 0; r < 8; ++r) rowacc[r] = FLT_MAX;

  const int nTiles  = (int)((Mb + 15) >> 4);
  const int nChunks = (nTiles + 3) >> 2;  // 4 tiles (64 target points) / chunk

  for (int cj = wave; cj < nChunks; cj += 8) {
    // Prefetch this wave's next chunk (global_prefetch_b8 path).
    {
      long long pfn = (long long)(cj + 8) * 64 + lm;
      if (pfn < Mb) __builtin_prefetch(t + (toff + pfn) * 3, 0, 1);
    }

#pragma unroll
    for (int k = 0; k < 4; ++k) {
      const int ct = cj * 4 + k;
      long long n  = (long long)ct * 16 + lm;
      bool      nv = (n < Mb);
      long long nc = nv ? n : (Mb - 1);  // clamp pad columns to a real point
      const float* tp = t + (toff + nc) * 3;
      float tx = tp[0], ty = tp[1], tz = tp[2];

      // B fragment: col N = lm, K = [-2tx,-2ty,-2tz,t2].
      v2f bb;
      bb.x = hi ? (-2.0f * tz) : (-2.0f * tx);
      bb.y = hi ? (tx * tx + ty * ty + tz * tz) : (-2.0f * ty);

      v8f d = __builtin_amdgcn_wmma_f32_16x16x4_f32(
          false, a, false, bb, (short)0, c, false, false);

      float cm = FLT_MAX;  // partial column min over this half's 8 rows
#pragma unroll
      for (int r = 0; r < 8; ++r) {
        float dr  = fmaxf(d[r], 0.0f);
        rowacc[r] = fminf(rowacc[r], nv ? dr : FLT_MAX);  // mask pad cols
        cm        = fminf(cm, dr);  // pad rows are ~3e36, never win
      }
      // Both halves hit the same address; hardware serializes same-address
      // lanes, so the combined result is the full column min.
      if (nv)
        atomicMin(&d2bits[toff + n], __float_as_int(cm));  // non-neg f32
    }
  }

  // Row min: reduce across the 16 column lanes within each half (cold path).
#pragma unroll
  for (int r = 0; r < 8; ++r) {
    float v = rowacc[r];
    v = fminf(v, __shfl_xor(v, 1, 32));
    v = fminf(v, __shfl_xor(v, 2, 32));
    v = fminf(v, __shfl_xor(v, 4, 32));
    v = fminf(v, __shfl_xor(v, 8, 32));
    if (lm == 0) {
      long long grow = (long long)rt * 16 + r + 8 * hi;
      if (grow < Nb)
        atomicMin(&d1bits[poff + grow], __float_as_int(v));
    }
  }
}

// ---------------------------------------------------------------------------
// Phase 2: per-batch sqrt + mean both directions, 0.5*(m1+m2)/B into out[0].
// ---------------------------------------------------------------------------
__global__ __launch_bounds__(256) void chamfer_finish(
    const int* __restrict__ d1bits, const int* __restrict__ d2bits,
    const long long* __restrict__ nums, const long long* __restrict__ dnums,
    int B, float* __restrict__ out) {
  const int b = blockIdx.x;
  long long poff = 0, toff = 0;
  for (int i = 0; i < b; ++i) { poff += nums[i]; toff += dnums[i]; }
  const long long Nb = nums[b], Mb = dnums[b];

  float s1 = 0.0f, s2 = 0.0f;
  for (long long i = threadIdx.x; i < Nb; i += blockDim.x)
    s1 += sqrtf(__int_as_float(d1bits[poff + i]));
  for (long long i = threadIdx.x; i < Mb; i += blockDim.x)
    s2 += sqrtf(__int_as_float(d2bits[toff + i]));

  __shared__ float sh[256];
  sh[threadIdx.x] = 0.5f * (s1 / (float)Nb + s2 / (float)Mb);
  __syncthreads();
  for (int s = 128; s > 0; s >>= 1) {
    if (threadIdx.x < s) sh[threadIdx.x] += sh[threadIdx.x + s];
    __syncthreads();
  }
  if (threadIdx.x == 0) atomicAdd(out, sh[0] / (float)B);
}

// ---------------------------------------------------------------------------
extern "C" void kernel_launch(void* const* d_in, const int* in_sizes, int n_in,
                              void* d_out, int out_size, void* d_ws,
                              size_t ws_size, hipStream_t stream) {
  const float*     p     = (const float*)d_in[0];
  const float*     t     = (const float*)d_in[1];
  const long long* nums  = (const long long*)d_in[2];  // int64 per reference
  const long long* dnums = (const long long*)d_in[3];
  const int B = in_sizes[2];

  const long long totalN = in_sizes[0] / 3;
  const long long totalM = in_sizes[1] / 3;

  int*   d1bits = (int*)d_ws;       // [totalN] min d^2 bits (pred -> target)
  int*   d2bits = d1bits + totalN;  // [totalM] min d^2 bits (target -> pred)
  float* out    = (float*)d_out;

  const long long tot = totalN + totalM;
  chamfer_init<<<(int)((tot + 255) / 256), 256, 0, stream>>>(d1bits, tot, out);

  // Upper bound on sum of per-batch row tiles: totalN/16 + B padding blocks.
  const int tilesUB = (int)((totalN + 15) / 16) + B;
  chamfer_tiles<<<tilesUB, 256, 0, stream>>>(p, t, nums, dnums, B, d1bits,
                                             d2bits);

  chamfer_finish<<<B, 256, 0, stream>>>(d1bits, d2bits, nums, dnums, B, out);
}